// MemoryQueue_77146202571048
// MI455X (gfx1250) — compile-verified
//
#include <hip/hip_runtime.h>

typedef float v2f __attribute__((ext_vector_type(2)));
typedef float v8f __attribute__((ext_vector_type(8)));
typedef int   v4i __attribute__((vector_size(16)));
typedef __attribute__((address_space(1))) v4i v4i_glob;
typedef __attribute__((address_space(3))) v4i v4i_lds;

constexpr int Bn = 64;    // batch
constexpr int Ln = 1024;  // l dimension (one workgroup each)
constexpr int Mn = 256;   // queue rows per l
constexpr int Fn = 256;   // feature dim (K of the GEMM)
constexpr int KC = 16;    // K-chunk staged in LDS (floats)
constexpr int NCH = Fn / KC;   // 16 chunks
constexpr int PAD = 20;   // padded LDS row stride (floats): 80B, 16B-aligned.
                          // 20*m mod 64 distinct for m=0..15 and ==0 mod 4 ->
                          // low-half b64 pairs on banks {0,1} mod 4, hi-half on
                          // {2,3} mod 4: conflict-free WMMA fragment reads.

#if defined(__has_builtin)
#if __has_builtin(__builtin_amdgcn_global_load_async_to_lds_b128)
#define ASYNC_LDS 1
#endif
#endif
#ifndef ASYNC_LDS
#define ASYNC_LDS 0
#endif

__device__ __forceinline__ void async_wait0() {
#if ASYNC_LDS
#if defined(__has_builtin) && __has_builtin(__builtin_amdgcn_s_wait_asynccnt)
  __builtin_amdgcn_s_wait_asynccnt(0);
#else
  asm volatile("s_wait_asynccnt 0x0" ::: "memory");
#endif
#endif
}

__device__ __forceinline__ void copy16_to_lds(const float* src, float* dst) {
#if ASYNC_LDS
  __builtin_amdgcn_global_load_async_to_lds_b128(
      (v4i_glob*)src, (v4i_lds*)dst, 0, 0);
#else
  float4 v = *(const float4*)src;
  *(float4*)dst = v;
#endif
}

// One workgroup per l, 8 waves:
//   wave w: b-tile bt = w>>1 (16 P-rows), m-half mh = w&1 (8 tiles of 16 m).
// Double-buffered pipeline: async-stage chunk c+1 into LDS while WMMAs consume
// chunk c; v_wmma_f32_16x16x4_f32 fed by conflict-free ds_load_b64 fragments.
__global__ __launch_bounds__(256) void mq_wmma_argmax_gather(
    const float* __restrict__ P,   // (B, L, F)
    const float* __restrict__ Q,   // (L, M, F)
    float* __restrict__ out) {     // (B, L, F)
  __shared__ float qs[2][Mn * PAD];   // 2 x 20,480 B
  __shared__ float ps[2][Bn * PAD];   // 2 x  5,120 B
  __shared__ float s_val2[2][Bn];
  __shared__ int   s_idx2[2][Bn];
  __shared__ int   s_idx[Bn];

  const int tid  = threadIdx.x;
  const int lane = tid & 31;
  const int wave = tid >> 5;
  const int bt   = wave >> 1;   // 0..3   b-tile
  const int mh   = wave & 1;    // 0..1   m half (m in [mh*128, mh*128+128))
  const int lm   = lane & 15;   // row/col within 16x16 tile
  const int lh   = lane >> 4;   // 0: K pair {0,1}, 1: K pair {2,3}
  const int l    = blockIdx.x;

  const float* Qb = Q + (size_t)l * Mn * Fn;

  // Staging assignment: thread covers (row = tid>>2 [+64*i], quad = tid&3);
  // 32 consecutive lanes cover 8 rows x 64B contiguous each.
  const int st_row  = tid >> 2;
  const int st_quad = tid & 3;

  auto stage = [&](int c, int bu) {
    const int kc = c * KC;
    #pragma unroll
    for (int i = 0; i < 4; ++i) {
      int row = st_row + 64 * i;
      copy16_to_lds(Qb + (size_t)row * Fn + kc + st_quad * 4,
                    &qs[bu][row * PAD + st_quad * 4]);
    }
    copy16_to_lds(P + ((size_t)st_row * Ln + l) * Fn + kc + st_quad * 4,
                  &ps[bu][st_row * PAD + st_quad * 4]);
  };

  v8f acc[8] = {};

  const int a_off = (bt * 16 + lm) * PAD + 2 * lh;   // fragment offsets (floats)
  const int b_off = (mh * 128 + lm) * PAD + 2 * lh;

  stage(0, 0);
  async_wait0();
  __syncthreads();

  #pragma unroll 1
  for (int c = 0; c < NCH; ++c) {
    const int cur = c & 1;
    if (c + 1 < NCH) stage(c + 1, cur ^ 1);  // prefetch next chunk (overlaps WMMA)

    const float* ap = &ps[cur][a_off];
    const float* bp = &qs[cur][b_off];
    #pragma unroll
    for (int kk = 0; kk < KC; kk += 4) {
      v2f a = *(const v2f*)(ap + kk);
      #pragma unroll
      for (int t = 0; t < 8; ++t) {
        v2f b = *(const v2f*)(bp + t * 16 * PAD + kk);
        acc[t] = __builtin_amdgcn_wmma_f32_16x16x4_f32(
            /*neg_a=*/false, a, /*neg_b=*/false, b,
            /*c_mod=*/(short)0, acc[t], /*reuse_a=*/false, /*reuse_b=*/false);
      }
    }

    async_wait0();    // chunk c+1 fully landed in LDS
    __syncthreads();  // all waves done with chunk c -> its buffer reusable
  }

  // Accumulator layout: vgpr r, lanes 0-15 -> b-row bt*16+r, lanes 16-31 -> +8,
  // column m = mh*128 + t*16 + lm.
  #pragma unroll
  for (int r = 0; r < 8; ++r) {
    float bv = acc[0][r];
    int   bi = mh * 128 + lm;
    #pragma unroll
    for (int t = 1; t < 8; ++t) {
      float v = acc[t][r];
      int   i = mh * 128 + t * 16 + lm;
      if (v > bv) { bv = v; bi = i; }   // strictly-greater keeps first max (low m)
    }
    // Butterfly over each 16-lane group (xor of bits 0..3 never crosses lane 16)
    #pragma unroll
    for (int off = 1; off < 16; off <<= 1) {
      float ov = __shfl_xor(bv, off, 32);
      int   oi = __shfl_xor(bi, off, 32);
      if (ov > bv || (ov == bv && oi < bi)) { bv = ov; bi = oi; }
    }
    if (lm == 0) {
      int b = bt * 16 + r + 8 * lh;
      s_val2[mh][b] = bv;
      s_idx2[mh][b] = bi;
    }
  }
  __syncthreads();

  // Combine the two m-halves (half-1 indices are larger, so ties pick half 0)
  if (tid < Bn) {
    float v0 = s_val2[0][tid];
    float v1 = s_val2[1][tid];
    s_idx[tid] = (v1 > v0) ? s_idx2[1][tid] : s_idx2[0][tid];
  }
  __syncthreads();

  // Gather winning rows: out[b, l, :] = Q[l, idx[b], :]  (64 rows x 1KB)
  {
    int b   = tid >> 2;
    int f0  = (tid & 3) * 64;
    int idx = s_idx[b];
    const float4* src = (const float4*)(Qb + (size_t)idx * Fn + f0);
    float4*       dst = (float4*)(out + ((size_t)b * Ln + l) * Fn + f0);
    #pragma unroll
    for (int j = 0; j < 16; ++j) dst[j] = src[j];
  }
}

extern "C" void kernel_launch(void* const* d_in, const int* in_sizes, int n_in,
                              void* d_out, int out_size, void* d_ws, size_t ws_size,
                              hipStream_t stream) {
  const float* P = (const float*)d_in[0];  // patch_features (B, L, F) fp32
  const float* Q = (const float*)d_in[1];  // queue (L, M, F) fp32
  float* out = (float*)d_out;              // (B, L, F) fp32
  mq_wmma_argmax_gather<<<dim3(Ln), dim3(256), 0, stream>>>(P, Q, out);
}